// WeightedSpatialSingularityRecalibration_81922206204087
// MI455X (gfx1250) — compile-verified
//
#include <hip/hip_runtime.h>
#include <hip/hip_bf16.h>
#include <float.h>

#define B_   16
#define H_   56
#define W_   56
#define C_   128
#define HW_  (H_*W_)
#define HWC_ (H_*W_*C_)
#define EPS_    1e-7f
#define BN_EPS_ 1e-3f
#define LN2_    0.69314718055994530942f

typedef __attribute__((ext_vector_type(2))) float v2f;
typedef __attribute__((ext_vector_type(8))) float v8f;

__device__ __forceinline__ float sigmoidf_(float v) { return 1.0f / (1.0f + expf(-v)); }

// ---------------------------------------------------------------------------
// Kernel 1: per-sample min/max over (H,W,C)
// ---------------------------------------------------------------------------
__global__ __launch_bounds__(256) void minmax_kernel(const float* __restrict__ x,
                                                     float* __restrict__ xmin,
                                                     float* __restrict__ xmax) {
    __shared__ float smin[256];
    __shared__ float smax[256];
    const int b = blockIdx.x;
    const float* xb = x + (size_t)b * HWC_;
    float mn = FLT_MAX, mx = -FLT_MAX;
    for (int i = threadIdx.x; i < HWC_; i += 256) {
        float v = xb[i];
        mn = fminf(mn, v);
        mx = fmaxf(mx, v);
    }
    smin[threadIdx.x] = mn;
    smax[threadIdx.x] = mx;
    __syncthreads();
    for (int o = 128; o > 0; o >>= 1) {
        if (threadIdx.x < (unsigned)o) {
            smin[threadIdx.x] = fminf(smin[threadIdx.x], smin[threadIdx.x + o]);
            smax[threadIdx.x] = fmaxf(smax[threadIdx.x], smax[threadIdx.x + o]);
        }
        __syncthreads();
    }
    if (threadIdx.x == 0) { xmin[b] = smin[0]; xmax[b] = smax[0]; }
}

// ---------------------------------------------------------------------------
// Summed-area table (57x57, zero row/col 0) for one (b,c) plane of xs in LDS.
// ---------------------------------------------------------------------------
__device__ __forceinline__ void build_sat(float* S, const float* __restrict__ x,
                                          int b, int c, float xmn, float inv) {
    const int tid = threadIdx.x;
    for (int i = tid; i < 57; i += 256) { S[i] = 0.0f; S[i * 57] = 0.0f; }
    for (int p = tid; p < HW_; p += 256) {
        int i = p / W_, j = p % W_;
        float xv = x[((size_t)(b * H_ + i) * W_ + j) * C_ + c];
        S[(i + 1) * 57 + (j + 1)] = (xv - xmn) * inv;
    }
    __syncthreads();
    if (tid < H_) {                       // inclusive row prefix, row tid+1
        float run = 0.0f;
        float* row = S + (tid + 1) * 57;
        for (int j = 1; j <= W_; ++j) { run += row[j]; row[j] = run; }
    }
    __syncthreads();
    if (tid < W_) {                       // inclusive col prefix, col tid+1
        float run = 0.0f;
        const int col = tid + 1;
        for (int i = 1; i <= H_; ++i) { run += S[i * 57 + col]; S[i * 57 + col] = run; }
    }
    __syncthreads();
}

// TF-SAME even-k box: rows [i-k/2+1, i+k/2], clamped (zero padding outside).
__device__ __forceinline__ float box_sum(const float* S, int i, int j, int khalf) {
    int r0 = i - khalf + 1; if (r0 < 0) r0 = 0;
    int r1 = i + khalf + 1; if (r1 > H_) r1 = H_;
    int c0 = j - khalf + 1; if (c0 < 0) c0 = 0;
    int c1 = j + khalf + 1; if (c1 > W_) c1 = W_;
    return S[r1 * 57 + c1] - S[r0 * 57 + c1] - S[r1 * 57 + c0] + S[r0 * 57 + c0];
}

// ---------------------------------------------------------------------------
// Kernel 2: per-(b,c) sum of alphas over the 56x56 plane (for squeeze)
// alpha = (lm3 - lm1 + 2*(lm4 - lm0)) / (10*ln2)   (xc coefficient for k=8 is 0)
// ---------------------------------------------------------------------------
__global__ __launch_bounds__(256) void alpha_sum_kernel(const float* __restrict__ x,
                                                        const float* __restrict__ xmin,
                                                        const float* __restrict__ xmax,
                                                        float* __restrict__ alpha_sum) {
    __shared__ float S[57 * 57];
    __shared__ float red[256];
    const int bc = blockIdx.x;
    const int b = bc >> 7, c = bc & 127;
    const float xmn = xmin[b];
    const float inv = 1.0f / (xmax[b] - xmn + EPS_);
    build_sat(S, x, b, c, xmn, inv);

    const float inv10ln2 = 1.0f / (10.0f * LN2_);
    float acc = 0.0f;
    for (int p = threadIdx.x; p < HW_; p += 256) {
        int i = p / W_, j = p % W_;
        float lm0 = logf(box_sum(S, i, j, 1)  + EPS_);
        float lm1 = logf(box_sum(S, i, j, 2)  + EPS_);
        float lm3 = logf(box_sum(S, i, j, 8)  + EPS_);
        float lm4 = logf(box_sum(S, i, j, 16) + EPS_);
        acc += (lm3 - lm1 + 2.0f * (lm4 - lm0)) * inv10ln2;
    }
    red[threadIdx.x] = acc;
    __syncthreads();
    for (int o = 128; o > 0; o >>= 1) {
        if (threadIdx.x < (unsigned)o) red[threadIdx.x] += red[threadIdx.x + o];
        __syncthreads();
    }
    if (threadIdx.x == 0) alpha_sum[bc] = red[0];
}

// ---------------------------------------------------------------------------
// Kernel 3: squeeze-excite with f32 WMMA (one wave32, EXEC all-ones).
//   hidden = relu(squeeze @ W1 + b1)   : 16x16 = [16,128]@[128,16]  (32 wmma)
//   excite = sigmoid(hidden @ W2 + b2) : 16x128 = [16,16]@[16,128]  (32 wmma)
// Operand layouts per CDNA5 ISA 7.12.2 (A 16x4: lane m=L%16, VGPR v half h -> K=v+2h;
// B 4x16: lane n=L%16, K=v+2h; C/D: lane n=L%16, M=v+8h).
// ---------------------------------------------------------------------------
__global__ __launch_bounds__(32) void se_wmma_kernel(const float* __restrict__ alpha_sum,
                                                     const float* __restrict__ W1,
                                                     const float* __restrict__ b1,
                                                     const float* __restrict__ W2,
                                                     const float* __restrict__ b2,
                                                     float* __restrict__ excite) {
    __shared__ float hid[16 * 16];
    const int lane = threadIdx.x;       // 0..31
    const int n    = lane & 15;         // column (and A-row) index
    const int h    = lane >> 4;         // half-wave select

    const float inv_hw = 1.0f / (float)HW_;

    // GEMM1: D = squeeze @ W1, K = 128 in steps of 4
    v8f acc = {};
    for (int k0 = 0; k0 < C_; k0 += 4) {
        v2f a, bm;
        const int ka = k0 + 2 * h;
        a.x  = alpha_sum[n * C_ + ka    ] * inv_hw;   // A[m=n][K=0+2h]
        a.y  = alpha_sum[n * C_ + ka + 1] * inv_hw;   // A[m=n][K=1+2h]
        bm.x = W1[(ka    ) * 16 + n];                 // B[K=0+2h][n]
        bm.y = W1[(ka + 1) * 16 + n];                 // B[K=1+2h][n]
        acc = __builtin_amdgcn_wmma_f32_16x16x4_f32(
                  false, a, false, bm, (short)0, acc, false, false);
    }
    // bias + relu, D layout -> LDS (row-major hidden[16][16])
    #pragma unroll
    for (int v = 0; v < 8; ++v) {
        int m = v + 8 * h;
        hid[m * 16 + n] = fmaxf(acc[v] + b1[n], 0.0f);
    }
    __syncthreads();

    // GEMM2: excite = hidden @ W2, 8 tiles of N=16, K=16
    for (int t = 0; t < 8; ++t) {
        v8f acc2 = {};
        for (int k0 = 0; k0 < 16; k0 += 4) {
            v2f a, bm;
            const int ka = k0 + 2 * h;
            a.x  = hid[n * 16 + ka    ];
            a.y  = hid[n * 16 + ka + 1];
            bm.x = W2[(ka    ) * C_ + t * 16 + n];
            bm.y = W2[(ka + 1) * C_ + t * 16 + n];
            acc2 = __builtin_amdgcn_wmma_f32_16x16x4_f32(
                       false, a, false, bm, (short)0, acc2, false, false);
        }
        #pragma unroll
        for (int v = 0; v < 8; ++v) {
            int m  = v + 8 * h;
            int cc = t * 16 + n;
            excite[m * C_ + cc] = sigmoidf_(acc2[v] + b2[cc]);
        }
    }
}

// ---------------------------------------------------------------------------
// Kernel 4: per-(b,c) plane — recompute SAT/log_m/alpha, cosine sim, BN gate,
// blend with excited xs, write output.
// ---------------------------------------------------------------------------
__global__ __launch_bounds__(256) void final_kernel(const float* __restrict__ x,
                                                    const float* __restrict__ kappa,
                                                    const float* __restrict__ gamma,
                                                    const float* __restrict__ beta,
                                                    const float* __restrict__ bn_mean,
                                                    const float* __restrict__ bn_var,
                                                    const float* __restrict__ xmin,
                                                    const float* __restrict__ xmax,
                                                    const float* __restrict__ excite,
                                                    float* __restrict__ out) {
    __shared__ float S[57 * 57];
    const int bc = blockIdx.x;
    const int b = bc >> 7, c = bc & 127;
    const float xmn = xmin[b];
    const float inv = 1.0f / (xmax[b] - xmn + EPS_);

    // centered (log_k + log_r) vector and its norm (uniform across threads)
    float uc[5];
    float vbar = 0.0f;
    #pragma unroll
    for (int s = 0; s < 5; ++s) {
        float sk  = sigmoidf_(kappa[s]);
        float slk = sigmoidf_(logf(sk + EPS_));
        uc[s] = slk + (float)(s + 1) * LN2_;
        vbar += uc[s];
    }
    vbar *= 0.2f;
    float nuc2 = 0.0f;
    #pragma unroll
    for (int s = 0; s < 5; ++s) { uc[s] -= vbar; nuc2 += uc[s] * uc[s]; }
    const float norm_uc = sqrtf(nuc2);

    // per-channel constants
    const float bn_scale = gamma[c] * rsqrtf(bn_var[c] + BN_EPS_);
    const float bn_mu    = bn_mean[c];
    const float bn_beta  = beta[c];
    const float exc      = excite[b * C_ + c];

    build_sat(S, x, b, c, xmn, inv);

    const float inv10ln2 = 1.0f / (10.0f * LN2_);
    for (int p = threadIdx.x; p < HW_; p += 256) {
        int i = p / W_, j = p % W_;
        float lm[5];
        lm[0] = logf(box_sum(S, i, j, 1)  + EPS_);
        lm[1] = logf(box_sum(S, i, j, 2)  + EPS_);
        lm[2] = logf(box_sum(S, i, j, 4)  + EPS_);
        lm[3] = logf(box_sum(S, i, j, 8)  + EPS_);
        lm[4] = logf(box_sum(S, i, j, 16) + EPS_);
        float mean_lm = 0.2f * (lm[0] + lm[1] + lm[2] + lm[3] + lm[4]);
        float alpha   = (lm[3] - lm[1] + 2.0f * (lm[4] - lm[0])) * inv10ln2;

        float dot_au = 0.0f, na2 = 0.0f;
        #pragma unroll
        for (int s = 0; s < 5; ++s) {
            float a = lm[s] - mean_lm;
            dot_au += a * uc[s];
            na2    += a * a;
        }
        float denom = sqrtf(na2) * fabsf(alpha) * norm_uc + EPS_;
        float cosv  = (alpha * dot_au) / denom;
        float sim   = 0.5f * (cosv + 1.0f);

        float an = bn_scale * (alpha - bn_mu) + bn_beta;

        size_t idx = ((size_t)(b * H_ + i) * W_ + j) * C_ + c;
        float xs = (x[idx] - xmn) * inv;
        out[idx] = sim * sigmoidf_(an) + (1.0f - sim) * xs * exc;
    }
}

// ---------------------------------------------------------------------------
// Launch
// ---------------------------------------------------------------------------
extern "C" void kernel_launch(void* const* d_in, const int* in_sizes, int n_in,
                              void* d_out, int out_size, void* d_ws, size_t ws_size,
                              hipStream_t stream) {
    const float* x       = (const float*)d_in[0];
    const float* kappa   = (const float*)d_in[1];
    const float* W1      = (const float*)d_in[2];
    const float* b1      = (const float*)d_in[3];
    const float* W2      = (const float*)d_in[4];
    const float* b2      = (const float*)d_in[5];
    const float* gamma   = (const float*)d_in[6];
    const float* beta    = (const float*)d_in[7];
    const float* bn_mean = (const float*)d_in[8];
    const float* bn_var  = (const float*)d_in[9];
    float* ws = (float*)d_ws;

    float* xmin      = ws;                 // 16
    float* xmax      = ws + 16;            // 16
    float* alpha_sum = ws + 32;            // 16*128
    float* excite    = ws + 32 + B_ * C_;  // 16*128
    float* out       = (float*)d_out;

    minmax_kernel   <<<B_,      256, 0, stream>>>(x, xmin, xmax);
    alpha_sum_kernel<<<B_ * C_, 256, 0, stream>>>(x, xmin, xmax, alpha_sum);
    se_wmma_kernel  <<<1,        32, 0, stream>>>(alpha_sum, W1, b1, W2, b2, excite);
    final_kernel    <<<B_ * C_, 256, 0, stream>>>(x, kappa, gamma, beta, bn_mean, bn_var,
                                                  xmin, xmax, excite, out);
}